// Linear_19421842112618
// MI455X (gfx1250) — compile-verified
//
#include <hip/hip_runtime.h>

// ---------------------------------------------------------------------------
// SASUT quantization-aware linear, MI455X (gfx1250, wave32).
//
//   out(64x8192) = inp_perm(64x8192) @ [noised(q_w) | fp_w](8192x8192)^T + bias
//
// Memory-bound: ~512 MB of weight+noise streamed once (floor ~22 us @ 23.3
// TB/s). GEMM MACs (8.6 GFLOP) routed through V_WMMA_F32_16X16X4_F32 so
// compute stays under the memory floor at full f32 precision.
// ---------------------------------------------------------------------------

typedef __attribute__((ext_vector_type(2))) float v2f;
typedef __attribute__((ext_vector_type(8))) float v8f;

#define TOKENS 64
#define INF    8192      // in_features (== K)
#define OUTF   8192      // out_features
#define QK     7936      // quantized (noised) K region
#define KC     64        // K chunk per LDS stage (divides both 7936 and 256)
#define PITCH  68        // LDS row pitch in floats: 272B = 16B-aligned rows,
                         // bank stride 68%64=4 -> conflict-free frag reads
#define QMAXF  7.0f      // 2^(4-1)-1

// ---------------------------------------------------------------------------
// Kernel 1: move the weight-column permutation onto the input.
//   inp_p[t, inv_col_perm[c]] = inp[t, c]
// ---------------------------------------------------------------------------
__global__ __launch_bounds__(256) void permute_input_kernel(
    const float* __restrict__ inp, const int* __restrict__ invp,
    float* __restrict__ inp_p)
{
    int idx = blockIdx.x * blockDim.x + threadIdx.x;   // 0 .. TOKENS*INF-1
    int c = idx & (INF - 1);
    int t = idx >> 13;                                 // idx / 8192
    int j = invp[c];
    inp_p[t * INF + j] = inp[idx];
}

// ---------------------------------------------------------------------------
// Kernel 2: fused weight-transform + skinny WMMA GEMM.
//   block(128 thr = 4 waves): tile = 64 tokens x 64 out-features
//   wave: 64 x 16 tile -> 4 accumulators (v8f) over 4 M-tiles of 16
// ---------------------------------------------------------------------------
__global__ __launch_bounds__(128) void sasut_gemm_kernel(
    const float* __restrict__ inp_p,   // 64 x 8192 (permuted)
    const float* __restrict__ q_w,     // 8192 x 7936
    const float* __restrict__ fp_w,    // 8192 x 256
    const float* __restrict__ alpha,   // 8192
    const float* __restrict__ bias,    // 8192
    const float* __restrict__ noise,   // 8192 x 7936
    float* __restrict__ out)           // 64 x 8192
{
    __shared__ float Alds[TOKENS * PITCH];   // 64 rows (tokens) x KC floats
    __shared__ float Blds[64 * PITCH];       // 64 rows (out-feat) x KC floats

    const int tid    = threadIdx.x;          // 0..127
    const int wave   = tid >> 5;             // 0..3
    const int lane   = tid & 31;
    const int nBlock = blockIdx.x * 64;      // block's out-feature base
    const int nWave  = nBlock + wave * 16;   // wave's 16 out-features

    v8f acc[4] = {v8f{}, v8f{}, v8f{}, v8f{}};

    const int nn = lane & 15;                // N index inside fragment
    const int kh = lane >> 4;                // K half-select (0: K0/K1, 1: K2/K3)

    for (int k0 = 0; k0 < INF; k0 += KC) {
        __syncthreads();   // LDS reuse fence vs previous chunk's compute

        // ---- stage A: 64x64 f32, coalesced float4 (128 thr x 8) ----
        #pragma unroll
        for (int i = 0; i < 8; ++i) {
            int idx = tid + i * 128;                 // float4 unit, 0..1023
            int row = idx >> 4;
            int kq  = (idx & 15) << 2;
            float4 v = *(const float4*)(inp_p + (size_t)row * INF + k0 + kq);
            *(float4*)(&Alds[row * PITCH + kq]) = v;
        }

        // ---- stage B: 64 out-feature rows x 64 K, transform fused ----
        if (k0 < QK) {
            // quantized region: w_noised = clamp-select with noise FMA
            #pragma unroll
            for (int i = 0; i < 8; ++i) {
                int idx = tid + i * 128;
                int row = idx >> 4;
                int kq  = (idx & 15) << 2;
                int n   = nBlock + row;
                float a  = alpha[n];
                float dh = 0.5f * a / QMAXF;          // 0.5 * delta
                float4 w  = *(const float4*)(q_w   + (size_t)n * QK + k0 + kq);
                float4 nz = *(const float4*)(noise + (size_t)n * QK + k0 + kq);
                float4 r;
                r.x = (w.x >= a) ? a : ((w.x <= -a) ? -a : fmaf(nz.x, dh, w.x));
                r.y = (w.y >= a) ? a : ((w.y <= -a) ? -a : fmaf(nz.y, dh, w.y));
                r.z = (w.z >= a) ? a : ((w.z <= -a) ? -a : fmaf(nz.z, dh, w.z));
                r.w = (w.w >= a) ? a : ((w.w <= -a) ? -a : fmaf(nz.w, dh, w.w));
                *(float4*)(&Blds[row * PITCH + kq]) = r;
            }
            // prefetch next chunk's weight/noise lines (latency hint)
            if (k0 + KC < QK) {
                int prow = tid >> 1;
                int pk   = (tid & 1) * 32;
                size_t off = (size_t)(nBlock + prow) * QK + (k0 + KC) + pk;
                __builtin_prefetch(q_w   + off, 0, 0);
                __builtin_prefetch(noise + off, 0, 0);
            }
        } else {
            // fp outlier region: pass-through
            #pragma unroll
            for (int i = 0; i < 8; ++i) {
                int idx = tid + i * 128;
                int row = idx >> 4;
                int kq  = (idx & 15) << 2;
                int n   = nBlock + row;
                float4 w = *(const float4*)(fp_w + (size_t)n * 256 + (k0 - QK) + kq);
                *(float4*)(&Blds[row * PITCH + kq]) = w;
            }
        }
        __syncthreads();

        // ---- compute: 16 K-steps of V_WMMA_F32_16X16X4_F32 ----
        // A frag (16x4): lanes 0-15 hold {K0,K1}, lanes 16-31 hold {K2,K3}
        // B frag (4x16): N striped across lanes, same K split across halves
        const float* Bbase = &Blds[(wave * 16 + nn) * PITCH + 2 * kh];
        const float* Abase = &Alds[nn * PITCH + 2 * kh];
        #pragma unroll
        for (int kk = 0; kk < KC; kk += 4) {
            v2f b = *(const v2f*)(Bbase + kk);        // ds_load_b64
            #pragma unroll
            for (int mt = 0; mt < 4; ++mt) {
                v2f a = *(const v2f*)(Abase + mt * 16 * PITCH + kk);
                acc[mt] = __builtin_amdgcn_wmma_f32_16x16x4_f32(
                    /*neg_a=*/false, a, /*neg_b=*/false, b,
                    /*c_mod=*/(short)0, acc[mt],
                    /*reuse_a=*/false, /*reuse_b=*/false);
            }
        }
    }

    // ---- epilogue: C/D layout -> global, add bias ----
    // lane L: N = L&15; VGPR r holds M = r + 8*(L>>4)
    int n = nWave + nn;
    float bv = bias[n];
    #pragma unroll
    for (int mt = 0; mt < 4; ++mt) {
        #pragma unroll
        for (int r = 0; r < 8; ++r) {
            int t = mt * 16 + r + 8 * kh;
            out[(size_t)t * OUTF + n] = acc[mt][r] + bv;
        }
    }
}

// ---------------------------------------------------------------------------
extern "C" void kernel_launch(void* const* d_in, const int* in_sizes, int n_in,
                              void* d_out, int out_size, void* d_ws, size_t ws_size,
                              hipStream_t stream) {
    const float* inp   = (const float*)d_in[0];
    const float* q_w   = (const float*)d_in[1];
    const float* fp_w  = (const float*)d_in[2];
    const float* alpha = (const float*)d_in[3];
    const float* bias  = (const float*)d_in[4];
    const float* noise = (const float*)d_in[5];
    const int*   invp  = (const int*)d_in[6];
    float* out   = (float*)d_out;
    float* inp_p = (float*)d_ws;   // needs TOKENS*INF*4 = 2 MB of scratch

    permute_input_kernel<<<dim3((TOKENS * INF) / 256), dim3(256), 0, stream>>>(
        inp, invp, inp_p);
    sasut_gemm_kernel<<<dim3(OUTF / 64), dim3(128), 0, stream>>>(
        inp_p, q_w, fp_w, alpha, bias, noise, out);
}